// Art_Metric_35777077576024
// MI455X (gfx1250) — compile-verified
//
#include <hip/hip_runtime.h>

#define B_  64
#define RN_ 60
#define P_  2
#define N_  2048
#define M_  1024
#define K_  64          // kNN ranks kept (after dropping self)
#define NT_ (M_ / 16)   // 64 column tiles
#define SPR 4           // sub-blocks per batch, rigid
#define SPP 2           // sub-blocks per (b,p), part

typedef __attribute__((ext_vector_type(2))) float v2f;
typedef __attribute__((ext_vector_type(8))) float v8f;

// ---- wave32 lane exchange via ds_bpermute (gfx1250 wave32) ----
__device__ __forceinline__ float bcastf(float v, int src) {
  return __uint_as_float((unsigned)__builtin_amdgcn_ds_bpermute(src << 2,
                          (int)__float_as_uint(v)));
}
__device__ __forceinline__ float xorf(float v, int mask) {
  return bcastf(v, ((int)(threadIdx.x & 31)) ^ mask);
}
__device__ __forceinline__ int xori(int v, int mask) {
  return __builtin_amdgcn_ds_bpermute(((((int)(threadIdx.x & 31)) ^ mask) << 2), v);
}

// ---- CDNA5 f32 WMMA, K=4 (A: 16x4, B: 4x16, C/D: 16x16 f32) ----
__device__ __forceinline__ v8f wmma_k4(v2f a, v2f b, v8f c) {
  return __builtin_amdgcn_wmma_f32_16x16x4_f32(false, a, false, b, (short)0, c,
                                               false, false);
}

__device__ __forceinline__ float dcd(float d, float k) { return 1.f - __expf(-k * d); }

__device__ float block_sum(float v, float* red) {
  int t = threadIdx.x;
  red[t] = v;
  __syncthreads();
  for (int s = 128; s > 0; s >>= 1) {
    if (t < s) red[t] += red[t + s];
    __syncthreads();
  }
  float r = red[0];
  __syncthreads();
  return r;
}

// ---------------------------------------------------------------- init
__global__ void k_init(float* wsf, unsigned* cm, int nf, int nc) {
  int i = blockIdx.x * blockDim.x + threadIdx.x;
  int stride = gridDim.x * blockDim.x;
  for (int j = i; j < nf; j += stride) wsf[j] = 0.f;
  for (int j = i; j < nc; j += stride) cm[j] = 0x7F800000u;  // +inf bits
}

// ---------------------------------------------------------------- rigid chamfer
// SPR blocks per batch; each covers a strided subset of the 128 x-strips.
// B-operands pre-swizzled in LDS as float4 {bx, by, |y|^2, 0}[half][j].
// Tile loop software-pipelined (prefetch tt+1 before processing tt) so the
// WMMA does not stall on the ds_load of its own tile. Column minima go to a
// block-local LDS array, merged into a per-batch global array at the end.
__global__ __launch_bounds__(256) void k_rigid(const float* __restrict__ S_align,
                                               const float* __restrict__ I_cano,
                                               unsigned* __restrict__ gcR,
                                               float* __restrict__ acc) {
  __shared__ float4 pk[2][M_];
  __shared__ unsigned cmin[M_];
  __shared__ float blk_fwd;
  const int blk = blockIdx.x, b = blk / SPR, sp = blk % SPR, t = threadIdx.x;
  if (t == 0) blk_fwd = 0.f;
  const float* yb = I_cano + (size_t)b * 3 * M_;
  for (int j = t; j < M_; j += 256) {
    float a0 = yb[j], a1 = yb[M_ + j], a2 = yb[2 * M_ + j];
    float yy = a0 * a0 + a1 * a1 + a2 * a2;
    pk[0][j] = make_float4(a0, a1, yy, 0.f);   // half 0: K=0,1
    pk[1][j] = make_float4(a2, 0.f, yy, 0.f);  // half 1: K=2,3 (pad)
    cmin[j] = 0x7F800000u;
  }
  __syncthreads();
  const int lane = t & 31, wave = t >> 5;
  const int half = lane >> 4, lj = lane & 15;
  const float4* pkh = pk[half];
  const float* xb = S_align + (size_t)b * 3 * N_;
  float fwd_acc = 0.f;

  for (int s = sp * 8 + wave; s < N_ / 16; s += 8 * SPR) {
    int i = s * 16 + lj;
    float x0 = xb[i], x1 = xb[N_ + i], x2 = xb[2 * N_ + i];
    float xx = x0 * x0 + x1 * x1 + x2 * x2;
    v2f a; a.x = half ? x2 : x0; a.y = half ? 0.f : x1;
    float xx8[8];
#pragma unroll
    for (int r = 0; r < 8; ++r) xx8[r] = bcastf(xx, r + 8 * half);
    float rmin[8];
#pragma unroll
    for (int r = 0; r < 8; ++r) rmin[r] = 3.0e38f;

    auto do_tile = [&](const float4& pv, int tt) {
      v2f bv; bv.x = pv.x; bv.y = pv.y;
      v8f c = {0.f, 0.f, 0.f, 0.f, 0.f, 0.f, 0.f, 0.f};
      c = wmma_k4(a, bv, c);
      float pmin = 3.0e38f;
#pragma unroll
      for (int r = 0; r < 8; ++r) {
        float d = fmaxf(fmaf(-2.f, c[r], xx8[r] + pv.z), 0.f);
        rmin[r] = fminf(rmin[r], d);
        pmin = fminf(pmin, d);
      }
      atomicMin(&cmin[tt * 16 + lj], __float_as_uint(pmin));
    };
    float4 pv = pkh[lj];
#pragma unroll 9
    for (int tt = 0; tt < NT_ - 1; ++tt) {   // prefetch next tile, process cur
      float4 pvn = pkh[(tt + 1) * 16 + lj];
      do_tile(pv, tt);
      pv = pvn;
    }
    do_tile(pv, NT_ - 1);

    float local = 0.f;
#pragma unroll
    for (int r = 0; r < 8; ++r) {    // row-min across 16 lanes of the half
      float v = rmin[r];
      v = fminf(v, xorf(v, 1));
      v = fminf(v, xorf(v, 2));
      v = fminf(v, xorf(v, 4));
      v = fminf(v, xorf(v, 8));
      local += dcd(v, 30.f);
    }
    if (lj == 0) fwd_acc += local;
  }
  if (lj == 0) atomicAdd(&blk_fwd, fwd_acc);
  __syncthreads();
  for (int j = t; j < M_; j += 256)            // merge to per-batch global min
    atomicMin(&gcR[(size_t)b * M_ + j], cmin[j]);
  if (t == 0) atomicAdd(acc, 5.f * blk_fwd / (float)(B_ * N_));
}

// ---------------------------------------------------------------- part chamfer
// SPP blocks per (b,p). 6-dim features padded to K=8 -> two chained f32 WMMAs
// per tile, software-pipelined LDS loads, global column-min merge.
__global__ __launch_bounds__(256) void k_part(const float* __restrict__ S_part,
                                              const float* __restrict__ S_color,
                                              const float* __restrict__ S_seg,
                                              const float* __restrict__ I_cano,
                                              const float* __restrict__ I_color,
                                              unsigned* __restrict__ gcP,
                                              float* __restrict__ acc) {
  __shared__ float4 pkA[2][M_];
  __shared__ float yssh[M_];
  __shared__ unsigned cmin[M_];
  __shared__ float blk_fwd;
  const int blk = blockIdx.x;
  const int sp = blk & (SPP - 1), p = (blk >> 1) & (P_ - 1), b = blk >> 2;
  const int t = threadIdx.x;
  if (t == 0) blk_fwd = 0.f;
  for (int j = t; j < M_; j += 256) {
    float g[6];
    float s2 = 0.f;
#pragma unroll
    for (int c = 0; c < 3; ++c) {
      g[c] = I_cano[((size_t)b * 3 + c) * M_ + j];
      g[3 + c] = 0.5f * I_color[((size_t)b * 3 + c) * M_ + j];
    }
#pragma unroll
    for (int c = 0; c < 6; ++c) s2 += g[c] * g[c];
    pkA[0][j] = make_float4(g[0], g[1], g[4], g[5]);  // half0: K0,1 | K4,5
    pkA[1][j] = make_float4(g[2], g[3], 0.f, 0.f);    // half1: K2,3 | K6,7 pad
    yssh[j] = s2;
    cmin[j] = 0x7F800000u;
  }
  __syncthreads();
  const int lane = t & 31, wave = t >> 5;
  const int half = lane >> 4, lj = lane & 15;
  const float4* pkh = pkA[half];
  float fwd_acc = 0.f;

  for (int s = sp * 8 + wave; s < N_ / 16; s += 8 * SPP) {
    int i = s * 16 + lj;
    float f[6];
#pragma unroll
    for (int c = 0; c < 3; ++c) {
      f[c] = S_part[(((size_t)b * P_ + p) * 3 + c) * N_ + i];
      f[3 + c] = 0.5f * S_color[((size_t)b * 3 + c) * N_ + i];
    }
    float xx = 0.f;
#pragma unroll
    for (int c = 0; c < 6; ++c) xx += f[c] * f[c];
    float w = S_seg[((size_t)b * P_ + p) * N_ + i];
    v2f a1; a1.x = half ? f[2] : f[0]; a1.y = half ? f[3] : f[1];  // K0..3
    v2f a2; a2.x = half ? 0.f : f[4];  a2.y = half ? 0.f : f[5];   // K4..7
    float xx8[8], w8[8];
#pragma unroll
    for (int r = 0; r < 8; ++r) {
      xx8[r] = bcastf(xx, r + 8 * half);
      w8[r]  = bcastf(w,  r + 8 * half);
    }
    float rmin[8];
#pragma unroll
    for (int r = 0; r < 8; ++r) rmin[r] = 3.0e38f;

    auto do_tile = [&](const float4& pv, float yy, int tt) {
      v2f b1; b1.x = pv.x; b1.y = pv.y;
      v2f b2; b2.x = pv.z; b2.y = pv.w;
      v8f c = {0.f, 0.f, 0.f, 0.f, 0.f, 0.f, 0.f, 0.f};
      c = wmma_k4(a1, b1, c);
      c = wmma_k4(a2, b2, c);
      float pmin = 3.0e38f;
#pragma unroll
      for (int r = 0; r < 8; ++r) {
        float d = fmaxf(fmaf(-2.f, c[r], xx8[r] + yy), 0.f);
        rmin[r] = fminf(rmin[r], d);
        pmin = fminf(pmin, d);
      }
      atomicMin(&cmin[tt * 16 + lj], __float_as_uint(pmin));
    };
    float4 pv = pkh[lj];
    float yy = yssh[lj];
#pragma unroll 9
    for (int tt = 0; tt < NT_ - 1; ++tt) {
      float4 pvn = pkh[(tt + 1) * 16 + lj];
      float yyn = yssh[(tt + 1) * 16 + lj];
      do_tile(pv, yy, tt);
      pv = pvn; yy = yyn;
    }
    do_tile(pv, yy, NT_ - 1);

    float local = 0.f;
#pragma unroll
    for (int r = 0; r < 8; ++r) {
      float v = rmin[r];
      v = fminf(v, xorf(v, 1));
      v = fminf(v, xorf(v, 2));
      v = fminf(v, xorf(v, 4));
      v = fminf(v, xorf(v, 8));
      local += dcd(v, 30.f) * w8[r];
    }
    if (lj == 0) fwd_acc += local;
  }
  if (lj == 0) atomicAdd(&blk_fwd, fwd_acc);
  __syncthreads();
  for (int j = t; j < M_; j += 256)
    atomicMin(&gcP[((size_t)b * P_ + p) * M_ + j], cmin[j]);
  if (t == 0) atomicAdd(acc, 5.f * blk_fwd / (float)(B_ * N_));
}

// ---------------------------------------------------------------- kNN gather
__global__ __launch_bounds__(256) void k_gather(const float* __restrict__ I_cano,
                                                float* __restrict__ gsum,
                                                float* __restrict__ gsq) {
  __shared__ float ys0[M_], ys1[M_], ys2[M_];
  __shared__ float dls[8][M_];
  __shared__ float ssum[K_], ssq[K_];
  const int blk = blockIdx.x, b = blk >> 7, rowbase = (blk & 127) * 8;
  const int t = threadIdx.x, lane = t & 31, wave = t >> 5;
  const float* yb = I_cano + (size_t)b * 3 * M_;
  for (int j = t; j < M_; j += 256) {
    ys0[j] = yb[j]; ys1[j] = yb[M_ + j]; ys2[j] = yb[2 * M_ + j];
  }
  if (t < K_) { ssum[t] = 0.f; ssq[t] = 0.f; }
  __syncthreads();
  const int m = rowbase + wave;
  const float x0 = ys0[m], x1 = ys1[m], x2 = ys2[m];
  float* dd = dls[wave];
  float lmin = 3.0e38f;
  int lidx = lane;
  for (int q = 0; q < 32; ++q) {
    int j = lane + 32 * q;
    float d0 = x0 - ys0[j], d1 = x1 - ys1[j], d2 = x2 - ys2[j];
    float d = d0 * d0 + d1 * d1 + d2 * d2;
    if (j == m) d = 3.0e38f;   // drop self (reference drops rank 0)
    dd[j] = d;
    if (d < lmin) { lmin = d; lidx = j; }
  }
  for (int k = 0; k < K_; ++k) {
    float v = lmin; int e = lidx;
#pragma unroll
    for (int msk = 1; msk < 32; msk <<= 1) {
      float v2 = xorf(v, msk);
      int e2 = xori(e, msk);
      if (v2 < v || (v2 == v && e2 < e)) { v = v2; e = e2; }
    }
    float r = sqrtf(fmaxf(v, 0.f));
    if (lane == 0) { atomicAdd(&ssum[k], r); atomicAdd(&ssq[k], r * r); }
    if ((e & 31) == lane) {       // winner removes its min and rescans
      dd[e] = 3.0e38f;
      lmin = 3.0e38f; lidx = lane;
      for (int q = 0; q < 32; ++q) {
        int j = lane + 32 * q;
        float d = dd[j];
        if (d < lmin) { lmin = d; lidx = j; }
      }
    }
  }
  __syncthreads();
  if (t < K_) {
    atomicAdd(&gsum[b * K_ + t], ssum[t]);
    atomicAdd(&gsq[b * K_ + t], ssq[t]);
  }
}

// ---------------------------------------------------------------- top-8 joint NN
__device__ float top8_dcd_sum(float* dj, int Np) {  // wave0 only; lane0 returns sum
  const int lane = threadIdx.x & 31;
  float s = 0.f;
  for (int k = 0; k < 8; ++k) {
    float lmin = 3.0e38f; int lidx = lane;
    for (int j = lane; j < Np; j += 32) {
      float d = dj[j];
      if (d < lmin) { lmin = d; lidx = j; }
    }
    float v = lmin; int e = lidx;
#pragma unroll
    for (int msk = 1; msk < 32; msk <<= 1) {
      float v2 = xorf(v, msk);
      int e2 = xori(e, msk);
      if (v2 < v || (v2 == v && e2 < e)) { v = v2; e = e2; }
    }
    if ((e & 31) == lane) dj[e] = 3.0e38f;
    if (lane == 0) s += dcd(v, 30.f);
  }
  return (lane == 0) ? s : 0.f;
}

// ---------------------------------------------------------------- small terms
// Runs after both chamfer kernels: also folds in the inverse-chamfer sums
// from the globally merged column minima.
__global__ __launch_bounds__(256) void k_small(const float* __restrict__ R_attn,
                                               const float* __restrict__ T_sel,
                                               const float* __restrict__ R_dist,
                                               const float* __restrict__ S_align,
                                               const float* __restrict__ S_joint,
                                               const float* __restrict__ S_seg,
                                               const float* __restrict__ I_cano,
                                               const float* __restrict__ I_joint,
                                               const float* __restrict__ I_drct,
                                               const float* __restrict__ I_angl,
                                               const float* __restrict__ I_svar,
                                               const float* __restrict__ I_seg,
                                               const unsigned* __restrict__ gcR,
                                               const unsigned* __restrict__ gcP,
                                               float* __restrict__ acc) {
  __shared__ float red[256];
  __shared__ float dj[N_];
  const int b = blockIdx.x, t = threadIdx.x;
  float contrib = 0.f;
  // rigid inverse chamfer: 1.25 * mean dcd120(colmin)
  for (int m = t; m < M_; m += 256)
    contrib += (1.25f / (float)(B_ * M_)) *
               dcd(__uint_as_float(gcR[(size_t)b * M_ + m]), 120.f);
  // part inverse chamfer, I_seg-weighted
  for (int p = 0; p < P_; ++p)
    for (int m = t; m < M_; m += 256)
      contrib += (1.25f / (float)(B_ * M_)) *
                 dcd(__uint_as_float(gcP[((size_t)b * P_ + p) * M_ + m]), 120.f) *
                 I_seg[((size_t)b * P_ + p) * M_ + m];
  // cano variance reg: 10 * mean dcd60
  for (int m = t; m < M_; m += 256)
    contrib += (10.f / (float)(B_ * M_)) * dcd(I_svar[b * M_ + m], 60.f);
  // attn + translation + joint regs
  if (t < RN_) contrib += R_attn[b * RN_ + t] * R_dist[b * RN_ + t] * (1.f / B_);
  if (t < 3) { float tv = T_sel[b * 3 + t]; contrib += tv * tv * (1.f / B_); }
  if (t == 0) {
    float d0 = I_drct[b * 3], d1 = I_drct[b * 3 + 1], d2 = I_drct[b * 3 + 2];
    float nrm = sqrtf(d0 * d0 + d1 * d1 + d2 * d2);
    contrib += (10.f / B_) * (nrm - 1.f) * (nrm - 1.f);
    float an = I_angl[b];
    contrib += (10.f / B_) * an * an;
    float j0 = I_joint[b * 3], j1 = I_joint[b * 3 + 1], j2 = I_joint[b * 3 + 2];
    contrib += (10.f / B_) * (j0 * j0 + j1 * j1 + j2 * j2);
  }
  // centroid penalty
  const float* yb = I_cano + (size_t)b * 3 * M_;
  float c0 = 0.f, c1 = 0.f, c2 = 0.f;
  for (int m = t; m < M_; m += 256) {
    c0 += yb[m]; c1 += yb[M_ + m]; c2 += yb[2 * M_ + m];
  }
  c0 = block_sum(c0, red); c1 = block_sum(c1, red); c2 = block_sum(c2, red);
  if (t == 0) {
    float inv = 1.f / (float)M_;
    contrib += (1.f / B_) * ((c0 * inv) * (c0 * inv) + (c1 * inv) * (c1 * inv) +
                             (c2 * inv) * (c2 * inv));
  }
  // prob hinges
  for (int p = 0; p < P_; ++p) {
    float sI = 0.f;
    for (int m = t; m < M_; m += 256) sI += I_seg[((size_t)b * P_ + p) * M_ + m];
    sI = block_sum(sI, red);
    if (t == 0) contrib += (10.f / (float)(B_ * P_)) * fmaxf(0.1f - sI / M_, 0.f);
    float sS = 0.f;
    for (int n = t; n < N_; n += 256) sS += S_seg[((size_t)b * P_ + p) * N_ + n];
    sS = block_sum(sS, red);
    if (t == 0) contrib += (10.f / (float)(B_ * P_)) * fmaxf(0.1f - sS / N_, 0.f);
  }
  // joint-closest (I side): top-8 NN of joint among I_cano points
  {
    float j0 = I_joint[b * 3], j1 = I_joint[b * 3 + 1], j2 = I_joint[b * 3 + 2];
    for (int m = t; m < M_; m += 256) {
      float d0 = j0 - yb[m], d1 = j1 - yb[M_ + m], d2 = j2 - yb[2 * M_ + m];
      dj[m] = d0 * d0 + d1 * d1 + d2 * d2;
    }
    __syncthreads();
    if (t < 32) contrib += (0.1f / (float)(B_ * 8)) * top8_dcd_sum(dj, M_);
    __syncthreads();
  }
  // joint-closest (S side): top-8 NN among S_align points
  {
    float j0 = S_joint[b * 3], j1 = S_joint[b * 3 + 1], j2 = S_joint[b * 3 + 2];
    const float* xb = S_align + (size_t)b * 3 * N_;
    for (int n = t; n < N_; n += 256) {
      float d0 = j0 - xb[n], d1 = j1 - xb[N_ + n], d2 = j2 - xb[2 * N_ + n];
      dj[n] = d0 * d0 + d1 * d1 + d2 * d2;
    }
    __syncthreads();
    if (t < 32) contrib += (0.1f / (float)(B_ * 8)) * top8_dcd_sum(dj, N_);
    __syncthreads();
  }
  float tot = block_sum(contrib, red);
  if (t == 0) atomicAdd(acc, tot);
}

// ---------------------------------------------------------------- finalize
__global__ __launch_bounds__(256) void k_final(const float* __restrict__ gsum,
                                               const float* __restrict__ gsq,
                                               const float* __restrict__ acc,
                                               float* __restrict__ out) {
  __shared__ float red[256];
  const int t = threadIdx.x;
  float v = 0.f;
  for (int i = t; i < B_ * K_; i += 256) {
    float s = gsum[i], q = gsq[i];
    v += (q - s * s * (1.f / (float)M_)) * (1.f / (float)(M_ - 1));  // ddof=1
  }
  float tot = block_sum(v, red);
  if (t == 0) out[0] = acc[0] + 200.f * tot / (float)(B_ * K_);
}

// ---------------------------------------------------------------- launch
extern "C" void kernel_launch(void* const* d_in, const int* in_sizes, int n_in,
                              void* d_out, int out_size, void* d_ws, size_t ws_size,
                              hipStream_t stream) {
  (void)in_sizes; (void)n_in; (void)out_size; (void)ws_size;
  const float* R_attn   = (const float*)d_in[0];
  const float* T_select = (const float*)d_in[1];
  const float* R_dist   = (const float*)d_in[2];
  const float* S_align  = (const float*)d_in[3];
  const float* S_part   = (const float*)d_in[4];
  const float* S_color  = (const float*)d_in[5];
  const float* S_joint  = (const float*)d_in[6];
  const float* S_seg    = (const float*)d_in[7];
  const float* I_cano   = (const float*)d_in[8];
  const float* I_color  = (const float*)d_in[9];
  const float* I_joint  = (const float*)d_in[10];
  const float* I_drct   = (const float*)d_in[11];
  const float* I_angl   = (const float*)d_in[12];
  const float* I_svar   = (const float*)d_in[13];
  const float* I_seg    = (const float*)d_in[14];

  float* ws        = (float*)d_ws;
  float* acc       = ws;                       // [1] scalar loss accumulator
  float* gsum      = ws + 1;                   // [B*K] per-rank kNN sums
  float* gsq       = ws + 1 + B_ * K_;         // [B*K] per-rank kNN sum-squares
  const int nf     = 1 + 2 * B_ * K_;
  unsigned* gcR    = (unsigned*)(ws + nf);     // [B*M] rigid column minima
  unsigned* gcP    = gcR + (size_t)B_ * M_;    // [B*P*M] part column minima
  const int nc     = B_ * M_ + B_ * P_ * M_;

  k_init<<<256, 256, 0, stream>>>(ws, gcR, nf, nc);
  k_rigid<<<B_ * SPR, 256, 0, stream>>>(S_align, I_cano, gcR, acc);
  k_part<<<B_ * P_ * SPP, 256, 0, stream>>>(S_part, S_color, S_seg, I_cano,
                                            I_color, gcP, acc);
  k_gather<<<B_ * (M_ / 8), 256, 0, stream>>>(I_cano, gsum, gsq);
  k_small<<<B_, 256, 0, stream>>>(R_attn, T_select, R_dist, S_align, S_joint,
                                  S_seg, I_cano, I_joint, I_drct, I_angl,
                                  I_svar, I_seg, gcR, gcP, acc);
  k_final<<<1, 256, 0, stream>>>(gsum, gsq, acc, (float*)d_out);
}